// WaveletTransform_53738630807635
// MI455X (gfx1250) — compile-verified
//
#include <hip/hip_runtime.h>

// Haar DWT, (16,64,256,256) f32 -> 4x (16,64,128,128) f32.
// Memory-bound streaming kernel (512 MB @ 23.3 TB/s ~ 22us). CDNA5 path:
// async global->LDS b128 DMA (ASYNCcnt) double-buffered per wave, wave32
// b128 coalescing, non-temporal b128 stores.

typedef float v4f __attribute__((ext_vector_type(4)));

#define HAAR_PLANE ((size_t)16 * 64 * 128 * 128)  // elements per output plane

__global__ __launch_bounds__(256) void haar_dwt_async_kernel(
    const float* __restrict__ x, float* __restrict__ out) {
  // 32 KB LDS: 8 waves x 2 buffers x 2048 B (one input row-pair per buffer)
  __shared__ __align__(16) float smem[8192];

  const int tid  = threadIdx.x;
  const int lane = tid & 31;
  const int wave = tid >> 5;
  const unsigned gw = blockIdx.x * 8u + (unsigned)wave;  // global wave id

  float* buf0 = &smem[wave * 1024];
  float* buf1 = &smem[wave * 1024 + 512];
  // Low 32 bits of a generic pointer into LDS == workgroup-relative LDS byte offset.
  const unsigned lds0 = (unsigned)(size_t)(buf0 + 4 * lane);
  const unsigned lds1 = (unsigned)(size_t)(buf1 + 4 * lane);

  // Issue 4 async b128 loads: one contiguous 2KB row-pair -> LDS buffer.
  // INST_OFFSET is added to BOTH the global and the LDS address (ISA 15.18),
  // so one per-lane base (+16B*lane) plus offsets 0/512/1024/1536 tiles it.
  auto issue = [&](unsigned t, unsigned ldsb) {
    const unsigned r  = gw * 4u + t;       // row task
    const unsigned i  = r & 127u;          // output row within plane
    const unsigned bc = r >> 7;            // fused batch*channel index
    const char* g = (const char*)x + (size_t)bc * 262144 /*256*256*4B*/
                                   + (size_t)i * 2048   /*row pair 2i,2i+1*/
                                   + 16 * lane;
    asm volatile(
        "global_load_async_to_lds_b128 %0, %1, off\n\t"
        "global_load_async_to_lds_b128 %0, %1, off offset:512\n\t"
        "global_load_async_to_lds_b128 %0, %1, off offset:1024\n\t"
        "global_load_async_to_lds_b128 %0, %1, off offset:1536"
        :: "v"(ldsb), "v"(g) : "memory");
  };

  // Consume one staged row-pair: lane L -> output cols 4L..4L+3.
  auto work = [&](unsigned t, const float* b) {
    const float* p = b + 8 * lane;
    v4f a0 = *(const v4f*)(p);        // row 2i,   cols 8L..8L+3
    v4f a1 = *(const v4f*)(p + 4);    // row 2i,   cols 8L+4..8L+7
    v4f r0 = *(const v4f*)(p + 256);  // row 2i+1, cols 8L..8L+3
    v4f r1 = *(const v4f*)(p + 260);  // row 2i+1, cols 8L+4..8L+7

    v4f s0 = {a0.x + a0.y, a0.z + a0.w, a1.x + a1.y, a1.z + a1.w};  // W-sums row0
    v4f d0 = {a0.x - a0.y, a0.z - a0.w, a1.x - a1.y, a1.z - a1.w};  // W-diffs row0
    v4f s1 = {r0.x + r0.y, r0.z + r0.w, r1.x + r1.y, r1.z + r1.w};  // W-sums row1
    v4f d1 = {r0.x - r0.y, r0.z - r0.w, r1.x - r1.y, r1.z - r1.w};  // W-diffs row1

    v4f ll = 0.5f * (s0 + s1);  // fL x fL
    v4f lh = 0.5f * (d0 + d1);  // fL(H) x fH(W)
    v4f hl = 0.5f * (s0 - s1);  // fH(H) x fL(W)
    v4f hh = 0.5f * (d0 - d1);  // fH x fH

    const unsigned r  = gw * 4u + t;
    const unsigned i  = r & 127u;
    const unsigned bc = r >> 7;
    float* o = out + ((size_t)bc * 128 + i) * 128 + 4 * lane;
    __builtin_nontemporal_store(ll, (v4f*)(o));
    __builtin_nontemporal_store(lh, (v4f*)(o + HAAR_PLANE));
    __builtin_nontemporal_store(hl, (v4f*)(o + 2 * HAAR_PLANE));
    __builtin_nontemporal_store(hh, (v4f*)(o + 3 * HAAR_PLANE));
  };

  // 4-stage, 2-deep software pipeline. Async loads retire in order, so
  // ASYNCcnt <= 4 guarantees the older stage's 4 loads have landed in LDS.
  issue(0, lds0);
  issue(1, lds1);
  asm volatile("s_wait_asynccnt 0x4" ::: "memory");
  work(0, buf0);
  issue(2, lds0);
  asm volatile("s_wait_asynccnt 0x4" ::: "memory");
  work(1, buf1);
  issue(3, lds1);
  asm volatile("s_wait_asynccnt 0x4" ::: "memory");
  work(2, buf0);
  asm volatile("s_wait_asynccnt 0x0" ::: "memory");
  work(3, buf1);
}

extern "C" void kernel_launch(void* const* d_in, const int* in_sizes, int n_in,
                              void* d_out, int out_size, void* d_ws, size_t ws_size,
                              hipStream_t stream) {
  const float* x = (const float*)d_in[0];  // (16,64,256,256) f32
  float* out = (float*)d_out;              // 4 planes of (16,64,128,128) f32
  (void)in_sizes; (void)n_in; (void)out_size; (void)d_ws; (void)ws_size;

  // 131072 output rows total; 8 waves/block x 4 rows/wave = 32 rows/block.
  dim3 grid(4096), block(256);
  haar_dwt_async_kernel<<<grid, block, 0, stream>>>(x, out);
}